// DeformableHead_7267084664917
// MI455X (gfx1250) — compile-verified
//
#include <hip/hip_runtime.h>
#include <hip/hip_bf16.h>
#include <math.h>

// ---------------------------------------------------------------------------
// Problem constants (from reference)
// ---------------------------------------------------------------------------
#define BB      2
#define TT      4
#define C_IN    128
#define HH      64
#define WW      64
#define HWPIX   4096          // 64*64
#define DM      256
#define NQ      300
#define NP      8
#define PS      3
#define NL      6
#define KPATCH  2304          // PS*PS*DM

typedef __attribute__((ext_vector_type(2))) float v2f;
typedef __attribute__((ext_vector_type(8))) float v8f;

// ---------------------------------------------------------------------------
// Activation helper
// ---------------------------------------------------------------------------
template<int ACT>
__device__ __forceinline__ float actf(float v) {
    if (ACT == 1) return v > 0.f ? v : 0.f;          // relu
    if (ACT == 2) return tanhf(v);                   // tanh
    if (ACT == 3) return 1.f / (1.f + expf(-v));     // sigmoid
    return v;
}

// ---------------------------------------------------------------------------
// Generic fp32 WMMA GEMM:  O = act(alpha * X @ W^T + bias)
//   X indexed  X[b*xbs + row*xsr + k*xsk]
//   W indexed  W[b*wbs + col*wsc + k*wsk]
//   O indexed  O[b*obs + row*osr + col*osc]
// One wave (32 lanes) per 16x16 tile. K must be a multiple of 4.
// Out-of-range rows/cols: clamped loads + guarded stores (EXEC stays all-1s).
// ---------------------------------------------------------------------------
template<int ACT, int ACC>
__global__ __launch_bounds__(32)
void wmma_gemm(const float* __restrict__ X, const float* __restrict__ Wt,
               const float* __restrict__ bias, float* __restrict__ O,
               int M, int N, int K,
               long xsr, long xsk, long wsc, long wsk,
               long osr, long osc, long xbs, long wbs, long obs,
               float alpha)
{
    const int lane = threadIdx.x;
    const int half = lane >> 4;        // 0: K0/K1, 1: K2/K3
    const int l16  = lane & 15;
    const int row0 = blockIdx.x << 4;
    const int col0 = blockIdx.y << 4;
    const int b    = blockIdx.z;

    const float* Xb = X  + (long)b * xbs;
    const float* Wb = Wt + (long)b * wbs;
    float*       Ob = O  + (long)b * obs;

    const int ra = (row0 + l16 < M) ? (row0 + l16) : (M - 1);
    const int ca = (col0 + l16 < N) ? (col0 + l16) : (N - 1);

    // per-lane base pointers; advance linearly in k
    const float* xp = Xb + (long)ra * xsr + (long)(half << 1) * xsk;
    const float* wp = Wb + (long)ca * wsc + (long)(half << 1) * wsk;
    const long xstep = 4 * xsk;
    const long wstep = 4 * wsk;

    v8f acc = {0.f, 0.f, 0.f, 0.f, 0.f, 0.f, 0.f, 0.f};

    #pragma unroll 4
    for (int k0 = 0; k0 < K; k0 += 4) {
        v2f av, bv;
        av[0] = xp[0];
        av[1] = xp[xsk];
        bv[0] = wp[0];
        bv[1] = wp[wsk];
        acc = __builtin_amdgcn_wmma_f32_16x16x4_f32(
                  false, av, false, bv, (short)0, acc, false, false);
        xp += xstep;
        wp += wstep;
    }

    const int ocol = col0 + l16;
    if (ocol < N) {
        #pragma unroll
        for (int r = 0; r < 8; ++r) {
            const int orow = row0 + r + (half << 3);
            if (orow < M) {
                float v = acc[r] * alpha;
                if (bias) v += bias[ocol];
                v = actf<ACT>(v);
                const long oi = (long)orow * osr + (long)ocol * osc;
                if (ACC) Ob[oi] += v; else Ob[oi] = v;
            }
        }
    }
}

static inline void launch_gemm(hipStream_t st, int act, int acc,
                               const float* X, const float* W, const float* bias,
                               float* O, int M, int N, int K,
                               long xsr, long xsk, long wsc, long wsk,
                               long osr, long osc,
                               long xbs, long wbs, long obs, int batches,
                               float alpha)
{
    dim3 g((M + 15) / 16, (N + 15) / 16, batches), blk(32);
#define LG(A, C) wmma_gemm<A, C><<<g, blk, 0, st>>>(X, W, bias, O, M, N, K, \
                     xsr, xsk, wsc, wsk, osr, osc, xbs, wbs, obs, alpha)
    if (acc)            { LG(0, 1); }
    else if (act == 1)  { LG(1, 0); }
    else if (act == 2)  { LG(2, 0); }
    else if (act == 3)  { LG(3, 0); }
    else                { LG(0, 0); }
#undef LG
}

// ---------------------------------------------------------------------------
// Implicit-GEMM 3x3 conv (stride 1, pad 1) on WMMA.
//   z  : (B, HWPIX, 256) pixel-major input
//   wsm: (256 co, 256 ci, 3, 3)   -> W element = wsm[co*2304 + ci*9 + p9]
//   f0 : (B, HWPIX, 256) output, + bias
// Loop order: patch tap p9 (fully unrolled; pad decision hoisted & branchless)
// then ci in steps of 4 (one WMMA per step, unrolled x4). EXEC never changes.
// A 16-row tile lies inside one image row (16 | 64), so h2/rowOK are uniform.
// ---------------------------------------------------------------------------
__global__ __launch_bounds__(32)
void conv3x3_wmma(const float* __restrict__ z, const float* __restrict__ wsm,
                  const float* __restrict__ bsm, float* __restrict__ f0)
{
    const int lane = threadIdx.x;
    const int half = lane >> 4;
    const int l16  = lane & 15;
    const int pix0 = blockIdx.x << 4;
    const int co0  = blockIdx.y << 4;
    const int b    = blockIdx.z;

    const float* zb = z + (long)b * HWPIX * DM;
    const int hh = pix0 >> 6;                 // uniform across the wave
    const int ww = (pix0 & 63) + l16;         // per-lane column
    const int co = co0 + l16;

    v8f acc = {0.f, 0.f, 0.f, 0.f, 0.f, 0.f, 0.f, 0.f};

    const int ci0 = half << 1;                // this half-wave's ci offset (0 or 2)

    #pragma unroll
    for (int p9 = 0; p9 < 9; ++p9) {
        const int ky = p9 / 3 - 1;            // compile-time constants
        const int kx = p9 % 3 - 1;
        const int h2 = hh + ky;
        const bool rowOK = (h2 >= 0) && (h2 < HH);
        const int w2 = ww + kx;
        const bool ok = rowOK && (w2 >= 0) && (w2 < WW);
        const int h2c = h2 < 0 ? 0 : (h2 >= HH ? HH - 1 : h2);
        const int w2c = w2 < 0 ? 0 : (w2 >= WW ? WW - 1 : w2);

        const float* ap = zb + (long)((h2c << 6) | w2c) * DM + ci0;   // per-lane base
        const float* bp = wsm + (long)co * KPATCH + ci0 * 9 + p9;

        #pragma unroll 4
        for (int ci = 0; ci < DM; ci += 4) {
            v2f av, bv;
            const float a0 = ap[0];
            const float a1 = ap[1];
            av[0] = ok ? a0 : 0.f;            // branchless pad (v_cndmask)
            av[1] = ok ? a1 : 0.f;
            bv[0] = bp[0];
            bv[1] = bp[9];
            acc = __builtin_amdgcn_wmma_f32_16x16x4_f32(
                      false, av, false, bv, (short)0, acc, false, false);
            ap += 4;
            bp += 36;                          // 4 * 9
        }
    }

    float* fb = f0 + (long)b * HWPIX * DM;
    const float bv2 = bsm[co];
    #pragma unroll
    for (int r = 0; r < 8; ++r) {
        const int orow = pix0 + r + (half << 3);
        fb[(long)orow * DM + co] = acc[r] + bv2;
    }
}

// ---------------------------------------------------------------------------
// Stage-A weight fusion:
//   mean_t(conv_t(feat)) = sum_{t'} Weff[t'] feat[t'],  Weff from surviving taps
//   W2[d][t'*128+ci] = sum_co w_in[d][co] * 0.25 * sum_{kk in S(t')} w_tf[co][ci][kk]
//   S(0)={0,1}, S(1)=S(2)={0,1,2}, S(3)={1,2}
// ---------------------------------------------------------------------------
__global__ void prep_w2_kernel(const float* __restrict__ w_tf,
                               const float* __restrict__ w_in,
                               float* __restrict__ W2)
{
    int t = blockIdx.x * blockDim.x + threadIdx.x;   // 256*512
    if (t >= DM * (TT * C_IN)) return;
    const int d  = t >> 9;       // / 512
    const int k  = t & 511;
    const int tp = k >> 7;       // 0..3
    const int ci = k & 127;
    const int klo = (tp == 3) ? 1 : 0;
    const int khi = (tp == 0) ? 1 : 2;
    float s = 0.f;
    for (int co = 0; co < C_IN; ++co) {
        const float* wt = w_tf + ((long)co * C_IN + ci) * 3;
        float wsum = 0.f;
        for (int kk = klo; kk <= khi; ++kk) wsum += wt[kk];
        s += w_in[(long)d * C_IN + co] * wsum;
    }
    W2[t] = 0.25f * s;
}

__global__ void prep_bias2_kernel(const float* __restrict__ w_in,
                                  const float* __restrict__ b_tf,
                                  const float* __restrict__ b_in,
                                  float* __restrict__ bias2)
{
    int d = threadIdx.x;   // 256
    float s = b_in[d];
    for (int co = 0; co < C_IN; ++co) s += w_in[(long)d * C_IN + co] * b_tf[co];
    bias2[d] = s;
}

// x[b][d][pix] += bias2[d] + sine_pos(d, h, w)
__global__ void add_pos_kernel(float* __restrict__ x, const float* __restrict__ bias2)
{
    long t = (long)blockIdx.x * blockDim.x + threadIdx.x;
    if (t >= (long)BB * DM * HWPIX) return;
    const int pix = (int)(t & (HWPIX - 1));
    const int d   = (int)((t >> 12) & (DM - 1));
    const int h = pix >> 6, w = pix & 63;
    const float TWO_PI = 6.283185307179586f;
    const int   c    = (d < 128) ? d : d - 128;
    const float base = (d < 128) ? (float)(h + 1) : (float)(w + 1);
    const float n    = base / (64.f + 1e-6f) * TWO_PI;
    const float dimt = powf(10000.f, (float)(c >> 2) / 32.f);
    const float arg  = n / dimt;
    const float val  = (c & 1) ? cosf(arg) : sinf(arg);
    x[t] += bias2[d] + val;
}

__global__ void init_queries_kernel(const float* __restrict__ q_embed,
                                    const float* __restrict__ q_pos,
                                    float* __restrict__ qry)
{
    int t = blockIdx.x * blockDim.x + threadIdx.x;   // BB*NQ*DM
    if (t >= BB * NQ * DM) return;
    const int qd = t % (NQ * DM);
    qry[t] = q_embed[qd] + q_pos[qd];
}

__global__ void copy_kernel(const float* __restrict__ src, float* __restrict__ dst, int n)
{
    int t = blockIdx.x * blockDim.x + threadIdx.x;
    if (t < n) dst[t] = src[t];
}

// Bilinear patch sampling from f0 (B, HWPIX, 256) -> flat (4800, 2304)
// one block = one (b,q,p) row, 256 threads = channels
__global__ void sample_kernel(const float* __restrict__ f0,
                              const float* __restrict__ boxes,
                              const float* __restrict__ ro,
                              float* __restrict__ flat)
{
    const int r  = blockIdx.x;       // (b*NQ+q)*NP + p
    const int c  = threadIdx.x;      // channel
    const int p  = r & 7;
    const int bq = r >> 3;
    const int b  = bq / NQ;

    const float rox = ro[(long)bq * 16 + p * 2 + 0] * 0.5f;
    const float roy = ro[(long)bq * 16 + p * 2 + 1] * 0.5f;
    const float refx = fminf(fmaxf(boxes[(long)bq * 4 + 0] + rox, 0.f), 1.f);
    const float refy = fminf(fmaxf(boxes[(long)bq * 4 + 1] + roy, 0.f), 1.f);

    const float* fb = f0 + (long)b * HWPIX * DM;

    for (int oyi = 0; oyi < PS; ++oyi) {
        const float oy = (float)(oyi - 1);
        const float gy = refy * 2.f - 1.f + oy * (2.f / (float)HH);
        const float ys = fminf(fmaxf((gy + 1.f) * 0.5f * (float)(HH - 1), 0.f), (float)(HH - 1));
        const float y0f = floorf(ys);
        const float wy  = ys - y0f;
        const int y0 = (int)y0f;
        const int y1 = (y0 + 1 < HH) ? y0 + 1 : HH - 1;
        for (int oxi = 0; oxi < PS; ++oxi) {
            const float ox = (float)(oxi - 1);
            const float gx = refx * 2.f - 1.f + ox * (2.f / (float)WW);
            const float xs = fminf(fmaxf((gx + 1.f) * 0.5f * (float)(WW - 1), 0.f), (float)(WW - 1));
            const float x0f = floorf(xs);
            const float wx  = xs - x0f;
            const int x0 = (int)x0f;
            const int x1 = (x0 + 1 < WW) ? x0 + 1 : WW - 1;

            const float v00 = fb[(long)((y0 << 6) + x0) * DM + c];
            const float v01 = fb[(long)((y0 << 6) + x1) * DM + c];
            const float v10 = fb[(long)((y1 << 6) + x0) * DM + c];
            const float v11 = fb[(long)((y1 << 6) + x1) * DM + c];
            const float val = v00 * (1.f - wx) * (1.f - wy) + v01 * wx * (1.f - wy)
                            + v10 * (1.f - wx) * wy         + v11 * wx * wy;
            flat[(long)r * KPATCH + (oyi * PS + oxi) * DM + c] = val;
        }
    }
}

// kv[row][c] = mean_p pv[row*8+p][c]
__global__ void kv_mean_kernel(const float* __restrict__ pv, float* __restrict__ kv)
{
    const int row = blockIdx.x;      // 0..599
    const int c   = threadIdx.x;     // 0..255
    float s = 0.f;
    for (int p = 0; p < NP; ++p) s += pv[((long)row * NP + p) * DM + c];
    kv[(long)row * DM + c] = s * (1.f / (float)NP);
}

// row-wise softmax over 300 cols; attn stored (600, 300) contiguous
__global__ void softmax_kernel(float* __restrict__ attn)
{
    __shared__ float red[128];
    const int row = blockIdx.x;
    const int tid = threadIdx.x;     // 128 threads
    float* a = attn + (long)row * NQ;

    float m = -1e30f;
    for (int j = tid; j < NQ; j += 128) m = fmaxf(m, a[j]);
    red[tid] = m; __syncthreads();
    for (int s = 64; s > 0; s >>= 1) {
        if (tid < s) red[tid] = fmaxf(red[tid], red[tid + s]);
        __syncthreads();
    }
    m = red[0]; __syncthreads();

    float sum = 0.f;
    for (int j = tid; j < NQ; j += 128) {
        float e = expf(a[j] - m);
        a[j] = e;
        sum += e;
    }
    red[tid] = sum; __syncthreads();
    for (int s = 64; s > 0; s >>= 1) {
        if (tid < s) red[tid] += red[tid + s];
        __syncthreads();
    }
    const float inv = 1.f / red[0];
    for (int j = tid; j < NQ; j += 128) a[j] *= inv;
}

// boxes = clip(boxes + 0.1*tanh(delta - 0.5), 0, 1); delta stored (600,4)
__global__ void update_boxes_kernel(float* __restrict__ boxes, const float* __restrict__ delta)
{
    int t = blockIdx.x * blockDim.x + threadIdx.x;
    if (t >= BB * NQ * 4) return;
    float v = boxes[t] + 0.1f * tanhf(delta[t] - 0.5f);
    boxes[t] = fminf(fmaxf(v, 0.f), 1.f);
}

// ---------------------------------------------------------------------------
// Host driver
// ---------------------------------------------------------------------------
extern "C" void kernel_launch(void* const* d_in, const int* in_sizes, int n_in,
                              void* d_out, int out_size, void* d_ws, size_t ws_size,
                              hipStream_t stream)
{
    (void)in_sizes; (void)n_in; (void)out_size; (void)ws_size;

    const float* feat    = (const float*)d_in[0];
    const float* boxes0  = (const float*)d_in[1];
    const float* w_tf    = (const float*)d_in[2];
    const float* b_tf    = (const float*)d_in[3];
    const float* w_in    = (const float*)d_in[4];
    const float* b_in    = (const float*)d_in[5];
    const float* w_lat   = (const float*)d_in[6];   // use scale 0 only (others are dead)
    const float* b_lat   = (const float*)d_in[7];
    const float* w_sm    = (const float*)d_in[8];
    const float* b_sm    = (const float*)d_in[9];
    const float* q_embed = (const float*)d_in[10];
    const float* q_pos   = (const float*)d_in[11];
    const float* Wq      = (const float*)d_in[12];
    const float* bq      = (const float*)d_in[13];
    const float* Wo      = (const float*)d_in[14];
    const float* bo      = (const float*)d_in[15];
    const float* Wp1     = (const float*)d_in[16];
    const float* bp1     = (const float*)d_in[17];
    const float* Wp2     = (const float*)d_in[18];
    const float* bp2     = (const float*)d_in[19];
    const float* w_r1    = (const float*)d_in[20];
    const float* b_r1    = (const float*)d_in[21];
    const float* w_r2    = (const float*)d_in[22];
    const float* b_r2    = (const float*)d_in[23];
    const float* w_b1    = (const float*)d_in[24];
    const float* b_b1    = (const float*)d_in[25];
    const float* w_b2    = (const float*)d_in[26];
    const float* b_b2    = (const float*)d_in[27];
    const float* w_cls   = (const float*)d_in[28];
    const float* b_cls   = (const float*)d_in[29];
    float* out = (float*)d_out;

    // workspace carve (floats)
    float* p = (float*)d_ws;
    float* W2    = p; p += DM * (TT * C_IN);        // 256x512
    float* bias2 = p; p += DM;
    float* x     = p; p += (long)BB * DM * HWPIX;   // (B, 256, 4096) channel-major
    float* z     = p; p += (long)BB * HWPIX * DM;   // (B, 4096, 256) pixel-major
    float* f0    = p; p += (long)BB * HWPIX * DM;   // (B, 4096, 256) pixel-major
    float* qry   = p; p += BB * NQ * DM;            // (600, 256)
    float* bxs   = p; p += BB * NQ * 4;
    float* h1    = p; p += BB * NQ * DM;
    float* ro    = p; p += BB * NQ * 16;
    float* flat  = p; p += (long)BB * NQ * NP * KPATCH;   // (4800, 2304)
    float* pv1   = p; p += (long)BB * NQ * NP * DM;       // (4800, 256)
    float* pv    = p; p += (long)BB * NQ * NP * DM;
    float* kv    = p; p += BB * NQ * DM;
    float* qb    = p; p += BB * NQ * DM;
    float* attn  = p; p += BB * NQ * NQ;                  // (2, 300, 300)
    float* outb  = p; p += BB * NQ * DM;
    float* hb    = p; p += BB * NQ * DM;
    float* delta = p; p += BB * NQ * 4;

    const long fbs = (long)TT * C_IN * HWPIX;   // feat batch stride
    const long xbs = (long)DM * HWPIX;          // x/z/f0 batch stride

    // --- Stage A: fused temporal-conv + T-mean + input projection ----------
    prep_w2_kernel<<<(DM * TT * C_IN + 255) / 256, 256, 0, stream>>>(w_tf, w_in, W2);
    prep_bias2_kernel<<<1, DM, 0, stream>>>(w_in, b_tf, b_in, bias2);
    // x[b][d][pix] = sum_k W2[d][k] * feat[b][k][pix]
    launch_gemm(stream, 0, 0, W2, feat, nullptr, x,
                DM, HWPIX, TT * C_IN,
                /*x*/ TT * C_IN, 1, /*w*/ 1, HWPIX, /*o*/ HWPIX, 1,
                0, fbs, xbs, BB, 1.f);
    add_pos_kernel<<<(int)((BB * (long)DM * HWPIX + 255) / 256), 256, 0, stream>>>(x, bias2);

    // --- FPN scale 0 only (scales 2 and 4 never reach the output) ---------
    // lateral 1x1:  z[b][pix][o] = sum_c x[b][c][pix] * w_lat0[o][c] + b_lat0
    launch_gemm(stream, 0, 0, x, w_lat, b_lat, z,
                HWPIX, DM, DM,
                /*x*/ 1, HWPIX, /*w*/ DM, 1, /*o*/ DM, 1,
                xbs, 0, xbs, BB, 1.f);
    // smooth 3x3 conv (implicit GEMM on WMMA)
    conv3x3_wmma<<<dim3(HWPIX / 16, DM / 16, BB), 32, 0, stream>>>(z, w_sm, b_sm, f0);

    // --- Decoder init ------------------------------------------------------
    init_queries_kernel<<<(BB * NQ * DM + 255) / 256, 256, 0, stream>>>(q_embed, q_pos, qry);
    copy_kernel<<<(BB * NQ * 4 + 255) / 256, 256, 0, stream>>>(boxes0, bxs, BB * NQ * 4);

    const int M6 = BB * NQ;        // 600
    for (int l = 0; l < NL; ++l) {
        // reference-point MLP
        launch_gemm(stream, 1, 0, qry, w_r1, b_r1, h1, M6, DM, DM,
                    DM, 1, DM, 1, DM, 1, 0, 0, 0, 1, 1.f);
        launch_gemm(stream, 2, 0, h1, w_r2, b_r2, ro, M6, 16, DM,
                    DM, 1, DM, 1, 16, 1, 0, 0, 0, 1, 1.f);
        // deformable sampling -> flat (4800, 2304)
        sample_kernel<<<BB * NQ * NP, DM, 0, stream>>>(f0, bxs, ro, flat);
        // patch MLP (dominant GEMM): 4800 x 256 x 2304
        launch_gemm(stream, 1, 0, flat, Wp1 + (long)l * DM * KPATCH, bp1 + l * DM, pv1,
                    BB * NQ * NP, DM, KPATCH,
                    KPATCH, 1, KPATCH, 1, DM, 1, 0, 0, 0, 1, 1.f);
        launch_gemm(stream, 0, 0, pv1, Wp2 + (long)l * DM * DM, bp2 + l * DM, pv,
                    BB * NQ * NP, DM, DM,
                    DM, 1, DM, 1, DM, 1, 0, 0, 0, 1, 1.f);
        kv_mean_kernel<<<M6, DM, 0, stream>>>(pv, kv);
        // attention
        launch_gemm(stream, 0, 0, qry, Wq + (long)l * DM * DM, bq + l * DM, qb,
                    M6, DM, DM, DM, 1, DM, 1, DM, 1, 0, 0, 0, 1, 1.f);
        launch_gemm(stream, 0, 0, qb, kv, nullptr, attn,
                    NQ, NQ, DM,
                    DM, 1, DM, 1, NQ, 1,
                    (long)NQ * DM, (long)NQ * DM, (long)NQ * NQ, BB, 0.0625f);
        softmax_kernel<<<M6, 128, 0, stream>>>(attn);
        launch_gemm(stream, 0, 0, attn, kv, nullptr, outb,
                    NQ, DM, NQ,
                    NQ, 1, 1, DM, DM, 1,
                    (long)NQ * NQ, (long)NQ * DM, (long)NQ * DM, BB, 1.f);
        // residual:  queries += outb @ Wo^T + bo
        launch_gemm(stream, 0, 1, outb, Wo + (long)l * DM * DM, bo + l * DM, qry,
                    M6, DM, DM, DM, 1, DM, 1, DM, 1, 0, 0, 0, 1, 1.f);
        // box head
        launch_gemm(stream, 1, 0, qry, w_b1, b_b1, hb, M6, DM, DM,
                    DM, 1, DM, 1, DM, 1, 0, 0, 0, 1, 1.f);
        launch_gemm(stream, 3, 0, hb, w_b2, b_b2, delta, M6, 4, DM,
                    DM, 1, DM, 1, 4, 1, 0, 0, 0, 1, 1.f);
        update_boxes_kernel<<<(BB * NQ * 4 + 255) / 256, 256, 0, stream>>>(bxs, delta);
    }

    // --- outputs: logits (600 floats) then boxes (2400 floats) ------------
    launch_gemm(stream, 0, 0, qry, w_cls, b_cls, out, M6, 1, DM,
                DM, 1, DM, 1, 1, 1, 0, 0, 0, 1, 1.f);
    copy_kernel<<<(BB * NQ * 4 + 255) / 256, 256, 0, stream>>>(bxs, out + M6, BB * NQ * 4);
}